// BEVDet_export_model_635655160580
// MI455X (gfx1250) — compile-verified
//
#include <hip/hip_runtime.h>
#include <hip/hip_bf16.h>

// ---------------------------------------------------------------------------
// BEVDet view-transform: 1x1 conv (GEMM 123x256 x 16896) -> softmax(59) ->
// depth (x) feat outer product -> scatter into 65537x64 BEV grid -> transpose
// to (1, 64, 256, 256).  Scatter-bandwidth bound (~310 MB total traffic).
// GEMM runs on V_WMMA_F32_16X16X4_F32 (fp32 matrix path, wave32).
// ---------------------------------------------------------------------------

#define DD   59       // depth bins
#define CC   64       // feature channels
#define OO   123      // D + C output channels
#define KK   256      // Cin
#define HW   2816     // 32*88  (divisible by 16)
#define NHW  16896    // 6*2816
#define NG   65536    // 256*256 grids
#define TS   16       // spatial tile (WMMA N dim)

typedef __attribute__((ext_vector_type(2))) float v2f;
typedef __attribute__((ext_vector_type(8))) float v8f;

// --------------------------- kernel 0: zero workspace ----------------------
__global__ __launch_bounds__(256) void zero_ws_kernel(float4* __restrict__ p, int n4) {
    int i = blockIdx.x * blockDim.x + threadIdx.x;
    int stride = gridDim.x * blockDim.x;
    float4 z = make_float4(0.f, 0.f, 0.f, 0.f);
    for (; i < n4; i += stride) p[i] = z;
}

// ------------- kernel 1: fused GEMM(WMMA f32) + softmax + scatter ----------
__global__ __launch_bounds__(256, 2) void fused_bev_kernel(
    const float* __restrict__ x_in,    // (6, 256, 32, 88)
    const float* __restrict__ W_dn,    // (123, 256) row-major
    const float* __restrict__ b_dn,    // (123)
    const int*   __restrict__ lidar,   // (6*59*2816)
    float*       __restrict__ ws)      // (65537, 64) scatter target
{
    __shared__ float ldsB[KK * TS];        // [k][s]      16 KB
    __shared__ float probT[TS * 60];       // [s][o<59]   (stride 60: -4s bank skew)
    __shared__ float tranT[TS * 68];       // [s][c<64]   (stride 68: 16B-aligned rows, 4s bank skew)
    __shared__ float smax[TS];
    __shared__ float srsum[TS];

    const int tid  = threadIdx.x;
    const int wave = tid >> 5;             // 0..7  -> M-tile (channels 16w..16w+15)
    const int lane = tid & 31;
    const int half = lane >> 4;            // 0/1
    const int lq   = lane & 15;

    const int s0  = blockIdx.x * TS;       // global spatial base (tile never crosses image)
    const int n   = s0 / HW;
    const int hw0 = s0 - n * HW;
    const float* xbase = x_in + (size_t)n * KK * HW + hw0;

    // ---- stage B tile: thread t = channel t, 16 consecutive floats (64B aligned)
    {
        const float4* src = (const float4*)(xbase + (size_t)tid * HW);
        float4* dst = (float4*)&ldsB[tid * TS];
        dst[0] = src[0]; dst[1] = src[1]; dst[2] = src[2]; dst[3] = src[3];
    }
    __syncthreads();

    // ---- GEMM: D(16x16) += A(16x4) * B(4x16), K = 256 in 64 steps.
    // A lane layout: M = lane%16, K-pair = k + 2*half ; B lane: N = lane%16, same K split.
    // Padding rows (channel >= 123) load a clamped (valid) row unconditionally:
    // their garbage lands only in D rows that are discarded at the store below,
    // so no EXEC-divergent guard is needed in the hot loop.
    v8f acc = {};
    const int  orow = wave * 16 + lq;
    const int  lrow = (orow < OO) ? orow : (OO - 1);    // clamp for safe loads
    const float* Wrow = W_dn + (size_t)lrow * KK + half * 2;
    const float* Brow = &ldsB[half * 2 * TS + lq];

    // one-stage software pipeline: fetch k+4 fragments before k's WMMA
    v2f a, b;
    a.x = Wrow[0];          a.y = Wrow[1];
    b.x = Brow[0];          b.y = Brow[TS];
    #pragma unroll 8
    for (int k = 4; k < KK; k += 4) {
        v2f an, bn;
        an.x = Wrow[k];               an.y = Wrow[k + 1];
        bn.x = Brow[k * TS];          bn.y = Brow[(k + 1) * TS];
        acc = __builtin_amdgcn_wmma_f32_16x16x4_f32(
                  false, a, false, b, (short)0, acc, false, false);
        a = an; b = bn;
    }
    acc = __builtin_amdgcn_wmma_f32_16x16x4_f32(
              false, a, false, b, (short)0, acc, false, false);

    // ---- bias + write tile transposed to LDS.
    // D layout: VGPR r -> M = r + 8*half ; N = lane%16.
    #pragma unroll
    for (int r = 0; r < 8; ++r) {
        int o = wave * 16 + r + half * 8;
        int s = lq;
        if (o < OO) {
            float v = acc[r] + b_dn[o];
            if (o < DD) probT[s * 60 + o]        = v;
            else        tranT[s * 68 + (o - DD)] = v;
        }
    }
    __syncthreads();

    // ---- softmax over depth (59) per position
    if (tid < TS) {
        float m = -3.402823466e+38f;
        for (int d = 0; d < DD; ++d) m = fmaxf(m, probT[tid * 60 + d]);
        float sum = 0.f;
        for (int d = 0; d < DD; ++d) sum += __expf(probT[tid * 60 + d] - m);
        smax[tid]  = m;
        srsum[tid] = 1.0f / sum;
    }
    __syncthreads();
    {
        int s  = tid & 15;
        int d0 = tid >> 4;
        float m = smax[s], rs = srsum[s];
        for (int d = d0; d < DD; d += 16)
            probT[s * 60 + d] = __expf(probT[s * 60 + d] - m) * rs;
    }
    __syncthreads();

    // ---- scatter: each half-wave owns one position; 59 rows of 256B each.
    {
        const int s = wave * 2 + half;                 // 0..15
        const float4 t4 = *(const float4*)&tranT[s * 68 + 4 * lq];  // invariant feature slice
        const int idxbase = n * DD * HW + hw0 + s;     // lidar index for (n, d=0, hw0+s)
        for (int d = 0; d < DD; ++d) {
            int idx  = lidar[idxbase + d * HW];        // 0..65536 (65536 = trash row)
            float p  = probT[s * 60 + d];
            float4 v = make_float4(p * t4.x, p * t4.y, p * t4.z, p * t4.w);
            *(float4*)&ws[(size_t)idx * CC + 4 * lq] = v;
        }
    }
}

// ---------------- kernel 2: transpose ws(grid,64) -> out(64,grid) ----------
__global__ __launch_bounds__(256) void transpose_bev_kernel(
    const float* __restrict__ ws, float* __restrict__ out)
{
    __shared__ float tile[64 * 65];
    const int t  = threadIdx.x;
    const int g0 = blockIdx.x * 64;

    #pragma unroll
    for (int j = 0; j < 4; ++j) {
        int lin = j * 256 + t;                 // float4 index 0..1023
        int row = lin >> 4;                    // grid within tile
        int c4  = (lin & 15) << 2;             // channel base
        float4 v = *(const float4*)&ws[(size_t)(g0 + row) * CC + c4];
        tile[row * 65 + c4 + 0] = v.x;
        tile[row * 65 + c4 + 1] = v.y;
        tile[row * 65 + c4 + 2] = v.z;
        tile[row * 65 + c4 + 3] = v.w;
    }
    __syncthreads();

    #pragma unroll
    for (int j = 0; j < 4; ++j) {
        int c  = (t >> 4) + j * 16;            // channel 0..63
        int gq = (t & 15) << 2;                // grid offset within tile
        float4 v;
        v.x = tile[(gq + 0) * 65 + c];
        v.y = tile[(gq + 1) * 65 + c];
        v.z = tile[(gq + 2) * 65 + c];
        v.w = tile[(gq + 3) * 65 + c];
        *(float4*)&out[(size_t)c * NG + g0 + gq] = v;
    }
}

// ---------------------------------------------------------------------------
extern "C" void kernel_launch(void* const* d_in, const int* in_sizes, int n_in,
                              void* d_out, int out_size, void* d_ws, size_t ws_size,
                              hipStream_t stream) {
    const float* x_in  = (const float*)d_in[0];
    const float* W_dn  = (const float*)d_in[1];
    const float* b_dn  = (const float*)d_in[2];
    const int*   lidar = (const int*)d_in[3];
    // d_in[4] (bev_feat zeros) unused: we scatter into workspace instead.

    float* out = (float*)d_out;
    float* ws  = (float*)d_ws;   // needs (NG+1)*64*4 = ~16.8 MB

    const int n4 = ((NG + 1) * CC) / 4;                 // float4 count
    zero_ws_kernel<<<2048, 256, 0, stream>>>((float4*)ws, n4);
    fused_bev_kernel<<<NHW / TS, 256, 0, stream>>>(x_in, W_dn, b_dn, lidar, ws);
    transpose_bev_kernel<<<NG / 64, 256, 0, stream>>>(ws, out);
}